// GCN_16441134809864
// MI455X (gfx1250) — compile-verified
//
#include <hip/hip_runtime.h>
#include <hip/hip_bf16.h>

typedef __attribute__((ext_vector_type(2))) float v2f;
typedef __attribute__((ext_vector_type(8))) float v8f;

#define N_GRAPHS 128

// ---------------------------------------------------------------- degree prep
__global__ __launch_bounds__(256) void k_init_deg(float* deg, int n) {
  int i = blockIdx.x * blockDim.x + threadIdx.x;
  if (i < n) deg[i] = 1.0f;  // +1 self loop
}

__global__ __launch_bounds__(256) void k_deg_edges(const int* __restrict__ dst,
                                                   float* __restrict__ deg, int E) {
  int e = blockIdx.x * blockDim.x + threadIdx.x;
  if (e < E) atomicAdd(&deg[dst[e]], 1.0f);
}

__global__ __launch_bounds__(256) void k_rsqrt_inplace(float* deg, int n) {
  int i = blockIdx.x * blockDim.x + threadIdx.x;
  if (i < n) deg[i] = rsqrtf(deg[i]);
}

// ---------------------------------------------------------------- WMMA GEMM
// H[M,N] = X[M,K] @ W[K,N], fp32 via V_WMMA_F32_16X16X4_F32.
// One wave -> 16-row strip x full N. Requires M%16==0, K%4==0, N%16==0.
template <int K, int N>
__global__ __launch_bounds__(256) void k_gemm_wmma(const float* __restrict__ X,
                                                   const float* __restrict__ W,
                                                   float* __restrict__ H, int M) {
  const int gwave = (blockIdx.x * blockDim.x + threadIdx.x) >> 5;
  const int lane  = threadIdx.x & 31;
  const int rowBase = gwave * 16;
  if (rowBase >= M) return;  // whole wave exits together -> EXEC all ones for WMMA

  const int m  = lane & 15;        // A row / B col / D col
  const int kh = (lane >> 4) * 2;  // K sub-offset: 0 (lanes 0-15) or 2 (lanes 16-31)
  constexpr int NT = N / 16;

  v8f acc[NT];
#pragma unroll
  for (int t = 0; t < NT; ++t) acc[t] = v8f{};

  for (int k = 0; k < K; k += 4) {
    // A 16x4 f32 fragment: VGPR0 = K=kh, VGPR1 = K=kh+1
    const float* xr = X + (size_t)(rowBase + m) * K + k + kh;
    v2f a; a.x = xr[0]; a.y = xr[1];
#pragma unroll
    for (int t = 0; t < NT; ++t) {
      // B 4x16 f32 fragment: lane = column, VGPR0 = row k+kh, VGPR1 = row k+kh+1
      const float* wr = W + (size_t)(k + kh) * N + t * 16 + m;
      v2f b; b.x = wr[0]; b.y = wr[N];
      acc[t] = __builtin_amdgcn_wmma_f32_16x16x4_f32(
          /*neg_a=*/false, a, /*neg_b=*/false, b,
          /*c_mod=*/(short)0, acc[t], /*reuse_a=*/false, /*reuse_b=*/false);
    }
  }

  // D 16x16 f32: VGPR v holds row (v + 8*(lane>=16)), col = lane&15
  const int mhi = (lane < 16) ? 0 : 8;
#pragma unroll
  for (int t = 0; t < NT; ++t) {
    float* out = H + (size_t)rowBase * N + t * 16 + (lane & 15);
#pragma unroll
    for (int v = 0; v < 8; ++v) out[(size_t)(v + mhi) * N] = acc[t][v];
  }
}

// ---------------------------------------------------------------- edge scatter
// agg[dst] += h[src] * dinv[src]*dinv[dst]; one thread per (edge, 4 features)
template <int F>
__global__ __launch_bounds__(256) void k_scatter(const int* __restrict__ src,
                                                 const int* __restrict__ dst,
                                                 const float* __restrict__ dinv,
                                                 const float* __restrict__ h,
                                                 float* __restrict__ agg, int E) {
  constexpr int CH = F / 4;
  unsigned t = blockIdx.x * blockDim.x + threadIdx.x;
  unsigned total = (unsigned)E * CH;
  if (t >= total) return;
  int e = (int)(t / CH);
  int c = (int)(t % CH);
  int s = src[e], d = dst[e];
  float coef = dinv[s] * dinv[d];
  const float4 hv = *(const float4*)(h + (size_t)s * F + c * 4);
  float* out = agg + (size_t)d * F + c * 4;
  atomicAdd(out + 0, hv.x * coef);
  atomicAdd(out + 1, hv.y * coef);
  atomicAdd(out + 2, hv.z * coef);
  atomicAdd(out + 3, hv.w * coef);
}

// --------------------------------------------- self-loop + bias (+ relu) fuse
template <int F, bool RELU>
__global__ __launch_bounds__(256) void k_finish(const float* __restrict__ h,
                                                const float* __restrict__ dinv,
                                                const float* __restrict__ b,
                                                float* __restrict__ agg, int M) {
  unsigned t = blockIdx.x * blockDim.x + threadIdx.x;
  if (t >= (unsigned)M * F) return;
  int i = (int)(t / F);
  int f = (int)(t % F);
  float di = dinv[i];
  float v = agg[t] + h[t] * di * di + b[f];
  agg[t] = RELU ? fmaxf(v, 0.0f) : v;
}

// ---------------------------------------------------------------- mean pool
__global__ __launch_bounds__(256) void k_pool(const float* __restrict__ h,
                                              const int* __restrict__ batch,
                                              float* __restrict__ sums,
                                              float* __restrict__ cnts, int M) {
  int i = blockIdx.x * (blockDim.x >> 5) + (threadIdx.x >> 5);
  int lane = threadIdx.x & 31;
  if (i >= M) return;
  int g = batch[i];
  const float4 v = *(const float4*)(h + (size_t)i * 128 + lane * 4);
  float* out = sums + (size_t)g * 128 + lane * 4;
  atomicAdd(out + 0, v.x);
  atomicAdd(out + 1, v.y);
  atomicAdd(out + 2, v.z);
  atomicAdd(out + 3, v.w);
  if (lane == 0) atomicAdd(&cnts[g], 1.0f);
}

// -------------------------------------------- head: mean, GEMV 128x14, lsm
__global__ __launch_bounds__(128) void k_head(const float* __restrict__ sums,
                                              const float* __restrict__ cnts,
                                              const float* __restrict__ Wl,
                                              const float* __restrict__ bl,
                                              float* __restrict__ out) {
  int g = threadIdx.x;  // 128 graphs
  float inv = 1.0f / fmaxf(cnts[g], 1.0f);
  float logits[14];
#pragma unroll
  for (int c = 0; c < 14; ++c) logits[c] = bl[c];
  for (int k = 0; k < 128; ++k) {
    float pk = sums[(size_t)g * 128 + k] * inv;
#pragma unroll
    for (int c = 0; c < 14; ++c) logits[c] = fmaf(pk, Wl[k * 14 + c], logits[c]);
  }
  float mx = logits[0];
#pragma unroll
  for (int c = 1; c < 14; ++c) mx = fmaxf(mx, logits[c]);
  float se = 0.0f;
#pragma unroll
  for (int c = 0; c < 14; ++c) se += expf(logits[c] - mx);
  float lse = mx + logf(se);
#pragma unroll
  for (int c = 0; c < 14; ++c) out[g * 14 + c] = logits[c] - lse;
}

// ---------------------------------------------------------------------------
static inline int cdiv(long long a, int b) { return (int)((a + b - 1) / b); }

extern "C" void kernel_launch(void* const* d_in, const int* in_sizes, int n_in,
                              void* d_out, int out_size, void* d_ws, size_t ws_size,
                              hipStream_t stream) {
  const float* x   = (const float*)d_in[0];
  const int*   ei  = (const int*)d_in[1];
  const int*   bat = (const int*)d_in[2];
  const float* W1  = (const float*)d_in[3];
  const float* b1  = (const float*)d_in[4];
  const float* W2  = (const float*)d_in[5];
  const float* b2  = (const float*)d_in[6];
  const float* W3  = (const float*)d_in[7];
  const float* b3  = (const float*)d_in[8];
  const float* Wl  = (const float*)d_in[9];
  const float* bl  = (const float*)d_in[10];
  float* out = (float*)d_out;

  const int M = in_sizes[0] / 20;  // 100000
  const int E = in_sizes[1] / 2;   // 3200000
  const int* src = ei;
  const int* dst = ei + E;

  // workspace layout (floats)
  float* dinv = (float*)d_ws;            // M
  float* bufA = dinv + 100032;           // M*128 (h1/h2/agg3)
  float* bufB = bufA + (size_t)M * 128;  // M*128 (agg1/h3)
  float* bufC = bufB + (size_t)M * 128;  // M*64  (agg2)
  float* sums = bufC + (size_t)M * 64;   // 128*128
  float* cnts = sums + N_GRAPHS * 128;   // 128

  // --- degrees -> dinv
  k_init_deg<<<cdiv(M, 256), 256, 0, stream>>>(dinv, M);
  k_deg_edges<<<cdiv(E, 256), 256, 0, stream>>>(dst, dinv, E);
  k_rsqrt_inplace<<<cdiv(M, 256), 256, 0, stream>>>(dinv, M);

  const int gemmBlocks = cdiv(M / 16, 8);  // 8 waves per block, 16 rows per wave

  // --- layer 1: 20 -> 64 (relu)
  k_gemm_wmma<20, 64><<<gemmBlocks, 256, 0, stream>>>(x, W1, bufA, M);
  hipMemsetAsync(bufB, 0, (size_t)M * 64 * sizeof(float), stream);
  k_scatter<64><<<cdiv((long long)E * 16, 256), 256, 0, stream>>>(src, dst, dinv, bufA, bufB, E);
  k_finish<64, true><<<cdiv((long long)M * 64, 256), 256, 0, stream>>>(bufA, dinv, b1, bufB, M);

  // --- layer 2: 64 -> 64 (relu)
  k_gemm_wmma<64, 64><<<gemmBlocks, 256, 0, stream>>>(bufB, W2, bufA, M);
  hipMemsetAsync(bufC, 0, (size_t)M * 64 * sizeof(float), stream);
  k_scatter<64><<<cdiv((long long)E * 16, 256), 256, 0, stream>>>(src, dst, dinv, bufA, bufC, E);
  k_finish<64, true><<<cdiv((long long)M * 64, 256), 256, 0, stream>>>(bufA, dinv, b2, bufC, M);

  // --- layer 3: 64 -> 128 (no relu)
  k_gemm_wmma<64, 128><<<gemmBlocks, 256, 0, stream>>>(bufC, W3, bufB, M);
  hipMemsetAsync(bufA, 0, (size_t)M * 128 * sizeof(float), stream);
  k_scatter<128><<<cdiv((long long)E * 32, 256), 256, 0, stream>>>(src, dst, dinv, bufB, bufA, E);
  k_finish<128, false><<<cdiv((long long)M * 128, 256), 256, 0, stream>>>(bufB, dinv, b3, bufA, M);

  // --- global mean pool + head
  hipMemsetAsync(sums, 0, (N_GRAPHS * 128 + N_GRAPHS) * sizeof(float), stream);
  k_pool<<<cdiv(M, 8), 256, 0, stream>>>(bufA, bat, sums, cnts, M);
  k_head<<<1, 128, 0, stream>>>(sums, cnts, Wl, bl, out);
}